// edgeConv_22797686407578
// MI455X (gfx1250) — compile-verified
//
#include <hip/hip_runtime.h>

typedef float v2f __attribute__((ext_vector_type(2)));
typedef float v8f __attribute__((ext_vector_type(8)));
typedef unsigned int u32x4 __attribute__((ext_vector_type(4)));
typedef int i32x4 __attribute__((ext_vector_type(4)));
typedef int i32x8 __attribute__((ext_vector_type(8)));

#define B_    8
#define FIN   64
#define NPTS  4096
#define FOUT  128
#define KNN   16

// workspace layout (float-element offsets), each region 256B aligned
#define WS_XT   0          // B*N*FIN           = 2097152 floats (8 MB)
#define WS_U    2097152    // B*N*FOUT          = 4194304 floats (16 MB)
#define WS_V    6291456    // B*N*FOUT          = 4194304 floats (16 MB)
#define WS_WT2  10485760   // FIN*256           = 16384 floats
#define WS_SQ   10502144   // B*N               = 32768 floats
#define WS_STAT 10534912   // 256 floats (sum, sumsq)
#define WS_SC   10535168   // 256 floats (scale, shift)
#define WS_IDX  10535424   // B*N*KNN ints      = 524288 ints

// ---------------------------------------------------------------- TDM helper
// LDS offset of a shared-memory object: flat shared-aperture addresses carry
// the wave-relative LDS byte offset in the low 32 bits.
__device__ __forceinline__ unsigned int lds_off(const void* p) {
  return (unsigned int)(uintptr_t)p;
}

// One TDM descriptor moving a contiguous 8 KB strip (2048 f32) global -> LDS.
// D# group0: count=1 | lds_addr | global_addr[56:0] | type=2 ("image").
// D# group1: data_size=4B, tensor_dim0=tile_dim0=2048, dim1=1, stride0=2048.
__device__ __forceinline__ void tdm_load_8kb(unsigned int lds_byte_off,
                                             const void* gptr) {
  unsigned long long ga = (unsigned long long)(uintptr_t)gptr;
  u32x4 g0;
  g0[0] = 1u;                                     // count=1 (valid user D#)
  g0[1] = lds_byte_off;                           // lds_addr
  g0[2] = (unsigned int)ga;                       // global_addr[31:0]
  g0[3] = (unsigned int)(ga >> 32) | 0x80000000u; // addr[56:32] | type=2<<30
  i32x8 g1;
  g1[0] = 0x00020000;                             // wg_mask=0, data_size=2 (4B)
  g1[1] = (int)(2048u << 16);                     // tensor_dim0[15:0]=2048
  g1[2] = (int)(1u << 16);                        // tensor_dim0 hi=0, tensor_dim1=1
  g1[3] = (int)(2048u << 16);                     // tensor_dim1 hi=0, tile_dim0=2048
  g1[4] = 1;                                      // tile_dim1=1, tile_dim2=0
  g1[5] = 2048;                                   // tensor_dim0_stride lo
  g1[6] = 0;                                      // stride0 hi, stride1 lo
  g1[7] = 0;                                      // stride1 hi
  i32x4 z4 = {0, 0, 0, 0};
  i32x8 z8 = {0, 0, 0, 0, 0, 0, 0, 0};
  // 6-arg (clang-23 / therock-10.0) form: groups 2,3 + extra group, cpol=0
  __builtin_amdgcn_tensor_load_to_lds(g0, g1, z4, z4, z8, 0);
}

// ---------------------------------------------------------------- transpose
__global__ __launch_bounds__(256) void transpose_kernel(const float* __restrict__ x,
                                                        float* __restrict__ xt) {
  int t = blockIdx.x * 256 + threadIdx.x;          // over B*FIN*N = 2^21
  int b = t >> 18;
  int r = t & ((1 << 18) - 1);
  int f = r >> 12;
  int n = r & 4095;
  xt[(size_t)b * NPTS * FIN + (size_t)n * FIN + f] = x[t];
}

// ---------------------------------------------------------------- |x|^2
__global__ __launch_bounds__(256) void sqnorm_kernel(const float* __restrict__ xt,
                                                     float* __restrict__ sq) {
  int bn = blockIdx.x * 256 + threadIdx.x;         // over B*N
  const float4* p = (const float4*)(xt + (size_t)bn * FIN);
  float s = 0.f;
#pragma unroll
  for (int i = 0; i < FIN / 4; ++i) {
    float4 q = p[i];
    s += q.x * q.x + q.y * q.y + q.z * q.z + q.w * q.w;
  }
  sq[bn] = s;
}

// ------------------------------------------------- fold W into (W1-W2 | W2)
__global__ __launch_bounds__(256) void prepw_kernel(const float* __restrict__ W,
                                                    float* __restrict__ WT2) {
  int t = blockIdx.x * 256 + threadIdx.x;          // 64*256 = 16384
  int f = t >> 8;
  int o = t & 255;
  float val;
  if (o < FOUT) val = W[o * 2 * FIN + f] - W[o * 2 * FIN + FIN + f];
  else          val = W[(o - FOUT) * 2 * FIN + FIN + f];
  WT2[t] = val;
}

__global__ void zero_kernel(float* p, int n) {
  int t = blockIdx.x * 256 + threadIdx.x;
  if (t < n) p[t] = 0.f;
}

// ---------------------------------------------------------------- kNN
// one wave per 16-row tile; TDM double-buffered candidate columns into LDS;
// WMMA f32 16x16x4 Gram tiles; per-lane sorted top-17 lists in LDS.
__global__ __launch_bounds__(32) void knn_kernel(const float* __restrict__ xt,
                                                 const float* __restrict__ sq,
                                                 int* __restrict__ idx_out) {
  __shared__ float lds_cols[2][32 * FIN];  // 2 x 8 KB double buffer (TDM dest)
  __shared__ float lds_tile[16 * 32];      // 2 KB: transposed Gram tile pair
  __shared__ float lds_sq[32];
  __shared__ float lds_bd[17 * 32];        // per-lane sorted dists (ascending)
  __shared__ int   lds_bi[17 * 32];

  const int lane = threadIdx.x;
  const int bt   = blockIdx.x >> 8;        // batch
  const int rt   = blockIdx.x & 255;       // row tile
  const int row0 = rt * 16;
  const int half = lane >> 4;
  const int l15  = lane & 15;

  const float* xb  = xt + (size_t)bt * NPTS * FIN;
  const float* sqb = sq + bt * NPTS;

  // A fragments for WMMA f32 16x16x4: VGPR0 = K {0|2}, VGPR1 = K {1|3}
  float areg[32];
  {
    const float* arow = xb + (size_t)(row0 + l15) * FIN;
#pragma unroll
    for (int s = 0; s < 16; ++s) {
      int k = 4 * s + (half ? 2 : 0);
      areg[2 * s]     = arow[k];
      areg[2 * s + 1] = arow[k + 1];
    }
  }

#pragma unroll
  for (int i = 0; i < 17; ++i) { lds_bd[i * 32 + lane] = 3.0e38f; lds_bi[i * 32 + lane] = 0; }

  const float rowsq = sqb[row0 + l15];
  float worst = 3.0e38f;

  // prologue: kick off DMA of tile 0 (32 columns x 64 f32, contiguous 8 KB)
  tdm_load_8kb(lds_off(&lds_cols[0][0]), xb);

  for (int mt = 0; mt < NPTS / 32; ++mt) {
    const int colbase = mt * 32;
    const int cur     = mt & 1;

    if (mt + 1 < NPTS / 32) {
      // prefetch next tile into the other buffer, then wait for current one
      tdm_load_8kb(lds_off(&lds_cols[cur ^ 1][0]),
                   xb + (size_t)(colbase + 32) * FIN);
      __builtin_amdgcn_s_wait_tensorcnt(1);   // in-order: tile mt complete
    } else {
      __builtin_amdgcn_s_wait_tensorcnt(0);
    }
    lds_sq[lane] = sqb[colbase + lane];
    __syncthreads();

#pragma unroll
    for (int t = 0; t < 2; ++t) {
      v8f c = {0.f, 0.f, 0.f, 0.f, 0.f, 0.f, 0.f, 0.f};
      const float* bcol = &lds_cols[cur][(t * 16 + l15) * FIN + (half ? 2 : 0)];
#pragma unroll
      for (int s = 0; s < 16; ++s) {
        v2f a;  a.x  = areg[2 * s]; a.y  = areg[2 * s + 1];
        v2f bb; bb.x = bcol[4 * s]; bb.y = bcol[4 * s + 1];
        c = __builtin_amdgcn_wmma_f32_16x16x4_f32(false, a, false, bb,
                                                  (short)0, c, false, false);
      }
      // C layout: VGPR r -> (row r|r+8, col lane&15); transpose into LDS
#pragma unroll
      for (int r = 0; r < 8; ++r) {
        int rr = r + (half ? 8 : 0);
        lds_tile[rr * 32 + t * 16 + l15] = c[r];
      }
    }
    __syncthreads();

    // each lane scans its half-tile (16 candidates) for its row
#pragma unroll
    for (int j = 0; j < 16; ++j) {
      int cloc = half * 16 + j;
      float d = rowsq + lds_sq[cloc] - 2.0f * lds_tile[l15 * 32 + cloc];
      if (d < worst) {
        int p = 16;
        while (p > 0 && lds_bd[(p - 1) * 32 + lane] > d) {
          lds_bd[p * 32 + lane] = lds_bd[(p - 1) * 32 + lane];
          lds_bi[p * 32 + lane] = lds_bi[(p - 1) * 32 + lane];
          --p;
        }
        lds_bd[p * 32 + lane] = d;
        lds_bi[p * 32 + lane] = colbase + cloc;
        worst = lds_bd[16 * 32 + lane];
      }
    }
    __syncthreads();
  }

  // merge the two half-lists per row; drop the smallest (self); emit 16
  if (lane < 16) {
    int ia = 0, ib = 0, written = 0;
    bool skipped = false;
    int* dst = idx_out + ((size_t)bt * NPTS + row0 + lane) * KNN;
    while (written < KNN) {
      float da = (ia < 17) ? lds_bd[ia * 32 + lane]      : 3.0e38f;
      float db = (ib < 17) ? lds_bd[ib * 32 + lane + 16] : 3.0e38f;
      int ci;
      if (da <= db) { ci = lds_bi[ia * 32 + lane];      ++ia; }
      else          { ci = lds_bi[ib * 32 + lane + 16]; ++ib; }
      if (!skipped) { skipped = true; continue; }
      dst[written++] = ci;      // batch-local point index, ascending distance
    }
  }
}

// ---------------------------------------------------------------- u,v GEMM
// per wave: 16 rows x 32 cols of (X^T @ [Wd|W2]^T), K=64, f32 WMMA
__global__ __launch_bounds__(32) void uv_gemm_kernel(const float* __restrict__ xt,
                                                     const float* __restrict__ WT2,
                                                     float* __restrict__ u,
                                                     float* __restrict__ v) {
  const int lane = threadIdx.x;
  const int half = lane >> 4;
  const int l15  = lane & 15;
  const int bt   = blockIdx.x / 2048;
  const int rem  = blockIdx.x % 2048;
  const int rt   = rem >> 3;
  const int ct   = rem & 7;
  const int row0 = rt * 16;
  const int col0 = ct * 32;

  const float* xb = xt + (size_t)bt * NPTS * FIN;
  float areg[32];
  const float* arow = xb + (size_t)(row0 + l15) * FIN;
#pragma unroll
  for (int s = 0; s < 16; ++s) {
    int k = 4 * s + (half ? 2 : 0);
    areg[2 * s]     = arow[k];
    areg[2 * s + 1] = arow[k + 1];
  }

  v8f c0 = {0.f, 0.f, 0.f, 0.f, 0.f, 0.f, 0.f, 0.f};
  v8f c1 = {0.f, 0.f, 0.f, 0.f, 0.f, 0.f, 0.f, 0.f};
#pragma unroll
  for (int s = 0; s < 16; ++s) {
    int k = 4 * s + (half ? 2 : 0);
    v2f a;  a.x  = areg[2 * s];                     a.y  = areg[2 * s + 1];
    v2f b0; b0.x = WT2[k * 256 + col0 + l15];       b0.y = WT2[(k + 1) * 256 + col0 + l15];
    v2f b1; b1.x = WT2[k * 256 + col0 + 16 + l15];  b1.y = WT2[(k + 1) * 256 + col0 + 16 + l15];
    c0 = __builtin_amdgcn_wmma_f32_16x16x4_f32(false, a, false, b0, (short)0, c0, false, false);
    c1 = __builtin_amdgcn_wmma_f32_16x16x4_f32(false, a, false, b1, (short)0, c1, false, false);
  }

  float* dst = (ct < 4) ? u : v;
  const int cbase = (ct < 4) ? col0 : col0 - 128;
#pragma unroll
  for (int r = 0; r < 8; ++r) {
    int rowg = row0 + r + (half ? 8 : 0);
    size_t base = ((size_t)bt * NPTS + rowg) * FOUT;
    dst[base + cbase + l15]      = c0[r];
    dst[base + cbase + 16 + l15] = c1[r];
  }
}

// ------------------------------------------------- BN statistics (sum,sumsq)
__global__ __launch_bounds__(128) void stats_kernel(const float* __restrict__ u,
                                                    const float* __restrict__ v,
                                                    const int* __restrict__ idx,
                                                    const float* __restrict__ bias,
                                                    float* __restrict__ stat) {
  __shared__ int lidx[64 * KNN];
  const int o     = threadIdx.x;
  const int pbase = blockIdx.x * 64;     // 64 points, same batch (4096%64==0)
  const int bt    = pbase >> 12;
#pragma unroll
  for (int q = 0; q < 8; ++q)
    lidx[q * 128 + o] = idx[(size_t)pbase * KNN + q * 128 + o];
  __syncthreads();

  const float bo = bias[o];
  const float* vb = v + ((size_t)bt << 12) * FOUT;
  float s = 0.f, ss = 0.f;
  for (int p = 0; p < 64; ++p) {
    float uo = u[((size_t)pbase + p) * FOUT + o] + bo;
#pragma unroll
    for (int j = 0; j < KNN; ++j) {
      float y = uo + vb[(size_t)lidx[p * KNN + j] * FOUT + o];
      s += y;
      ss += y * y;
    }
  }
  atomicAdd(&stat[o], s);
  atomicAdd(&stat[FOUT + o], ss);
}

// ------------------------------------------------- fold BN scale/shift+bias
__global__ void finalize_kernel(const float* __restrict__ stat,
                                const float* __restrict__ gamma,
                                const float* __restrict__ beta,
                                const float* __restrict__ bias,
                                float* __restrict__ sc) {
  int o = threadIdx.x;
  if (o < FOUT) {
    const float inv_cnt = 1.0f / (float)(B_ * NPTS * KNN);
    float mean  = stat[o] * inv_cnt;
    float var   = stat[FOUT + o] * inv_cnt - mean * mean;
    float scale = gamma[o] * rsqrtf(var + 1e-5f);
    sc[o]        = scale;
    sc[FOUT + o] = (bias[o] - mean) * scale + beta[o];   // bias folded in
  }
}

// ------------------------------------------------- normalize+ReLU+max, store
__global__ __launch_bounds__(128) void output_kernel(const float* __restrict__ u,
                                                     const float* __restrict__ v,
                                                     const int* __restrict__ idx,
                                                     const float* __restrict__ sc,
                                                     float* __restrict__ out) {
  __shared__ float tile[FOUT * 16];
  __shared__ int   lidx[16 * KNN];
  const int o     = threadIdx.x;
  const int pbase = blockIdx.x * 16;
  const int bt    = pbase >> 12;
  const int n0    = pbase & 4095;

  lidx[o]       = idx[(size_t)pbase * KNN + o];
  lidx[o + 128] = idx[(size_t)pbase * KNN + o + 128];
  __syncthreads();

  const float scale = sc[o];
  const float shift = sc[FOUT + o];
  const float* vb = v + ((size_t)bt << 12) * FOUT;
  for (int p = 0; p < 16; ++p) {
    float uo = u[((size_t)pbase + p) * FOUT + o];
    float m = -3.0e38f;
#pragma unroll
    for (int j = 0; j < KNN; ++j) {
      float t = (uo + vb[(size_t)lidx[p * KNN + j] * FOUT + o]) * scale + shift;
      m = fmaxf(m, t);
    }
    tile[o * 16 + p] = fmaxf(m, 0.f);     // relu(max_j) == max_j relu
  }
  __syncthreads();
  // coalesced transposed store: thread o writes 16 consecutive n
  float* dst = out + ((size_t)bt * FOUT + o) * NPTS + n0;
#pragma unroll
  for (int q = 0; q < 4; ++q) {
    float4 w;
    w.x = tile[o * 16 + q * 4 + 0];
    w.y = tile[o * 16 + q * 4 + 1];
    w.z = tile[o * 16 + q * 4 + 2];
    w.w = tile[o * 16 + q * 4 + 3];
    *(float4*)&dst[q * 4] = w;
  }
}

extern "C" void kernel_launch(void* const* d_in, const int* in_sizes, int n_in,
                              void* d_out, int out_size, void* d_ws, size_t ws_size,
                              hipStream_t stream) {
  const float* x     = (const float*)d_in[0];
  const float* W     = (const float*)d_in[1];
  const float* bias  = (const float*)d_in[2];
  const float* gamma = (const float*)d_in[3];
  const float* beta  = (const float*)d_in[4];
  // d_in[5] is k == 16, hardcoded

  float* ws   = (float*)d_ws;
  float* xt   = ws + WS_XT;
  float* u    = ws + WS_U;
  float* v    = ws + WS_V;
  float* WT2  = ws + WS_WT2;
  float* sq   = ws + WS_SQ;
  float* stat = ws + WS_STAT;
  float* sc   = ws + WS_SC;
  int*   idx  = (int*)(ws + WS_IDX);
  float* out  = (float*)d_out;

  transpose_kernel<<<8192, 256, 0, stream>>>(x, xt);
  sqnorm_kernel<<<128, 256, 0, stream>>>(xt, sq);
  prepw_kernel<<<64, 256, 0, stream>>>(W, WT2);
  zero_kernel<<<1, 256, 0, stream>>>(stat, 256);
  knn_kernel<<<B_ * (NPTS / 16), 32, 0, stream>>>(xt, sq, idx);
  uv_gemm_kernel<<<B_ * (NPTS / 16) * 8, 32, 0, stream>>>(xt, WT2, u, v);
  stats_kernel<<<B_ * NPTS / 64, 128, 0, stream>>>(u, v, idx, bias, stat);
  finalize_kernel<<<1, 128, 0, stream>>>(stat, gamma, beta, bias, sc);
  output_kernel<<<B_ * NPTS / 16, 128, 0, stream>>>(u, v, idx, sc, out);
}